// GRN_66949950210693
// MI455X (gfx1250) — compile-verified
//
#include <hip/hip_runtime.h>

// GIN inference for MI455X (gfx1250, wave32).
// All working buffers (~77MB) fit in the 192MB L2 -> L2-BW / L2-atomic bound.
// fp32 WMMA (V_WMMA_F32_16X16X4_F32) for all 128-wide GEMMs; epilogues fused.

#define N_NODES   50000
#define N_EDGES   800000
#define SD        128
#define N_GRAPHS  128
#define N_CLASSES 41
#define BN_EPS    1e-5f

typedef __attribute__((ext_vector_type(2))) float v2f;
typedef __attribute__((ext_vector_type(8))) float v8f;

#define EPI_RELU    0
#define EPI_RELU_BN 1

// x[n, c] = emb[node_ids[n], c]
__global__ void embed_kernel(const int* __restrict__ node_ids,
                             const float* __restrict__ emb,
                             float* __restrict__ x) {
    int tid = blockIdx.x * blockDim.x + threadIdx.x;
    if (tid >= N_NODES * SD) return;
    int n = tid >> 7;
    int c = tid & 127;
    x[tid] = emb[node_ids[n] * SD + c];
}

__global__ void copy4_kernel(const float4* __restrict__ src,
                             float4* __restrict__ dst, int n4) {
    int tid = blockIdx.x * blockDim.x + threadIdx.x;
    if (tid < n4) dst[tid] = src[tid];
}

// One wave32 per edge: lane l handles cols l, l+32, l+64, l+96 (coalesced).
// h (pre-initialized to x) accumulates x[src] into row dst -> h = x + agg.
__global__ void edge_scatter_kernel(const int* __restrict__ src,
                                    const int* __restrict__ dst,
                                    const float* __restrict__ x,
                                    float* __restrict__ h) {
    int gid = blockIdx.x * blockDim.x + threadIdx.x;
    int e = gid >> 5;
    if (e >= N_EDGES) return;
    int lane = gid & 31;
    const float* xs = x + src[e] * SD;
    float* hd = h + dst[e] * SD;
#pragma unroll
    for (int i = 0; i < 4; ++i)
        atomicAdd(&hd[lane + i * 32], xs[lane + i * 32]);  // non-returning f32 atomic
}

// out[M,128] = epilogue(A[M,128] @ W[128,128]^T + bias)
// Block = 256 threads (8 waves). Block handles 16 rows; wave w handles cols 16w..16w+15.
// K loop: 32 x V_WMMA_F32_16X16X4_F32 accumulations.
template <int EPI>
__global__ void gemm_wmma_kernel(const float* __restrict__ A,
                                 const float* __restrict__ W,
                                 const float* __restrict__ bias,
                                 const float* __restrict__ bn_gamma,
                                 const float* __restrict__ bn_beta,
                                 const float* __restrict__ bn_mean,
                                 const float* __restrict__ bn_var,
                                 float* __restrict__ out) {
    const int wave = threadIdx.x >> 5;
    const int lane = threadIdx.x & 31;
    const int m0   = blockIdx.x * 16;
    const int n0   = wave * 16;
    const int half = lane >> 4;   // 0: lanes 0-15, 1: lanes 16-31
    const int l15  = lane & 15;

    // A frag (16x4 f32): lane row = m0+l15; half 0 holds K(+0,+1), half 1 holds K(+2,+3)
    // B frag (4x16 f32): lane col = n0+l15; same K split. B = W^T -> read W[col][k..k+1].
    const int   koff = half * 2;
    const float* Ap = A + (m0 + l15) * SD + koff;
    const float* Wp = W + (n0 + l15) * SD + koff;

    v8f acc = {};
#pragma unroll
    for (int k = 0; k < SD; k += 4) {
        v2f a = *(const v2f*)(Ap + k);   // 8B-aligned float2
        v2f b = *(const v2f*)(Wp + k);
        acc = __builtin_amdgcn_wmma_f32_16x16x4_f32(
            /*neg_a=*/false, a, /*neg_b=*/false, b,
            /*c_mod=*/(short)0, acc, /*reuse_a=*/false, /*reuse_b=*/false);
    }

    // C/D layout: VGPR r -> row m0 + r + 8*half, col n0 + l15
    const int col = n0 + l15;
    float bv = bias[col];
    float s = 1.f, mu = 0.f, be = 0.f;
    if (EPI == EPI_RELU_BN) {
        s  = bn_gamma[col] * rsqrtf(bn_var[col] + BN_EPS);
        mu = bn_mean[col];
        be = bn_beta[col];
    }
#pragma unroll
    for (int r = 0; r < 8; ++r) {
        float z = acc[r] + bv;
        z = fmaxf(z, 0.f);
        if (EPI == EPI_RELU_BN) z = (z - mu) * s + be;
        out[(m0 + r + 8 * half) * SD + col] = z;
    }
}

__global__ void zero_kernel(float* __restrict__ p, int n) {
    int tid = blockIdx.x * blockDim.x + threadIdx.x;
    if (tid < n) p[tid] = 0.f;
}

__global__ void pool_kernel(const int* __restrict__ batch,
                            const float* __restrict__ x,
                            float* __restrict__ pooled) {
    int tid = blockIdx.x * blockDim.x + threadIdx.x;
    if (tid >= N_NODES * SD) return;
    int n = tid >> 7;
    int c = tid & 127;
    atomicAdd(&pooled[batch[n] * SD + c], x[tid]);
}

// out[g, j] = h[g,:] . fc2_W[j,:] + fc2_b[j]   (128 x 41, negligible cost)
__global__ void fc2_kernel(const float* __restrict__ h,
                           const float* __restrict__ W,
                           const float* __restrict__ b,
                           float* __restrict__ out) {
    int tid = blockIdx.x * blockDim.x + threadIdx.x;
    if (tid >= N_GRAPHS * N_CLASSES) return;
    int g = tid / N_CLASSES;
    int j = tid - g * N_CLASSES;
    const float* hp = h + g * SD;
    const float* wp = W + j * SD;
    float acc = 0.f;
#pragma unroll 8
    for (int k = 0; k < SD; ++k) acc = fmaf(hp[k], wp[k], acc);
    out[tid] = acc + b[j];
}

extern "C" void kernel_launch(void* const* d_in, const int* in_sizes, int n_in,
                              void* d_out, int out_size, void* d_ws, size_t ws_size,
                              hipStream_t stream) {
    const int*   node_ids = (const int*)d_in[0];
    const int*   edge_src = (const int*)d_in[1];             // edge_index[0]
    const int*   edge_dst = (const int*)d_in[1] + N_EDGES;   // edge_index[1]
    const int*   batch    = (const int*)d_in[2];
    const float* emb      = (const float*)d_in[3];
    const float* in_W1    = (const float*)d_in[4];
    const float* in_b1    = (const float*)d_in[5];
    const float* in_W2    = (const float*)d_in[6];
    const float* in_b2    = (const float*)d_in[7];
    const float* bn_g     = (const float*)d_in[8];
    const float* bn_b     = (const float*)d_in[9];
    const float* bn_m     = (const float*)d_in[10];
    const float* bn_v     = (const float*)d_in[11];
    const float* out_W1   = (const float*)d_in[12];
    const float* out_b1   = (const float*)d_in[13];
    const float* out_W2   = (const float*)d_in[14];
    const float* out_b2   = (const float*)d_in[15];
    const float* fc1_W    = (const float*)d_in[16];
    const float* fc1_b    = (const float*)d_in[17];
    const float* fc2_W    = (const float*)d_in[18];
    const float* fc2_b    = (const float*)d_in[19];
    float* out = (float*)d_out;

    const int FEAT = N_NODES * SD;  // 6.4M floats
    float* x      = (float*)d_ws;          // 25.6 MB
    float* h      = x + FEAT;              // 25.6 MB
    float* t      = h + FEAT;              // 25.6 MB
    float* pooled = t + FEAT;              // 64 KB
    float* hfc    = pooled + N_GRAPHS * SD;// 64 KB

    const int TPB = 256;
    const int gFeat = (FEAT + TPB - 1) / TPB;
    const int gGemm = N_NODES / 16;  // 50000 = 16*3125, exact

    // x = emb[node_ids]
    embed_kernel<<<gFeat, TPB, 0, stream>>>(node_ids, emb, x);

    // 5 shared "in" layers (ReLU+BN epilogue) + 1 "out" layer (ReLU epilogue)
    for (int layer = 0; layer < 6; ++layer) {
        const bool isOut = (layer == 5);
        const float* W1 = isOut ? out_W1 : in_W1;
        const float* b1 = isOut ? out_b1 : in_b1;
        const float* W2 = isOut ? out_W2 : in_W2;
        const float* b2 = isOut ? out_b2 : in_b2;

        // h = x, then h[dst] += x[src]  ->  h = x + segment_sum(x[src], dst)
        copy4_kernel<<<(FEAT / 4 + TPB - 1) / TPB, TPB, 0, stream>>>(
            (const float4*)x, (float4*)h, FEAT / 4);
        edge_scatter_kernel<<<(N_EDGES * 32 + TPB - 1) / TPB, TPB, 0, stream>>>(
            edge_src, edge_dst, x, h);

        // t = relu(h @ W1^T + b1)
        gemm_wmma_kernel<EPI_RELU><<<gGemm, TPB, 0, stream>>>(
            h, W1, b1, nullptr, nullptr, nullptr, nullptr, t);

        // x = bn(relu(t @ W2^T + b2))   (out layer: relu only)
        if (isOut)
            gemm_wmma_kernel<EPI_RELU><<<gGemm, TPB, 0, stream>>>(
                t, W2, b2, nullptr, nullptr, nullptr, nullptr, x);
        else
            gemm_wmma_kernel<EPI_RELU_BN><<<gGemm, TPB, 0, stream>>>(
                t, W2, b2, bn_g, bn_b, bn_m, bn_v, x);
    }

    // pooled[g] = sum_{batch[n]==g} x[n]
    zero_kernel<<<(N_GRAPHS * SD + TPB - 1) / TPB, TPB, 0, stream>>>(pooled, N_GRAPHS * SD);
    pool_kernel<<<gFeat, TPB, 0, stream>>>(batch, x, pooled);

    // hfc = relu(pooled @ fc1_W^T + fc1_b)  (M=128 -> 8 blocks, same WMMA kernel)
    gemm_wmma_kernel<EPI_RELU><<<N_GRAPHS / 16, TPB, 0, stream>>>(
        pooled, fc1_W, fc1_b, nullptr, nullptr, nullptr, nullptr, hfc);

    // out = hfc @ fc2_W^T + fc2_b  [128, 41]
    fc2_kernel<<<(N_GRAPHS * N_CLASSES + TPB - 1) / TPB, TPB, 0, stream>>>(
        hfc, fc2_W, fc2_b, out);
}